// binaryTreeLogicNet_14061722928009
// MI455X (gfx1250) — compile-verified
//
#include <hip/hip_runtime.h>

#define EPSF 1e-6f
#define NSZ 64            // INPUT_SIZE == NUM_LEAVES
#define NLAYERS 63
#define BATCH_TOTAL 262144
#define SINK_ITERS 20
#define ROWS_PER_BLOCK 128

typedef __attribute__((ext_vector_type(2))) float v2f;
typedef __attribute__((ext_vector_type(8))) float v8f;

// ---------------- Kernel 1: log-space Sinkhorn on 64x64 logits -> P (workspace)
__global__ __launch_bounds__(64) void btln_sinkhorn_kernel(const float* __restrict__ logits,
                                                           float* __restrict__ P) {
    __shared__ float l[64 * 65];          // pad 65 -> conflict-free column access
    const int t = threadIdx.x;            // 0..63
    for (int j = 0; j < 64; ++j) l[t * 65 + j] = logits[t * 64 + j];
    __syncthreads();
    for (int it = 0; it < SINK_ITERS; ++it) {
        // axis=1 (rows): thread t owns row t
        float m = -3.4e38f;
        for (int j = 0; j < 64; ++j) m = fmaxf(m, l[t * 65 + j]);
        float s = 0.f;
        for (int j = 0; j < 64; ++j) s += expf(l[t * 65 + j] - m);
        float lse = m + logf(s);
        for (int j = 0; j < 64; ++j) l[t * 65 + j] -= lse;
        __syncthreads();
        // axis=0 (cols): thread t owns column t
        m = -3.4e38f;
        for (int i = 0; i < 64; ++i) m = fmaxf(m, l[i * 65 + t]);
        s = 0.f;
        for (int i = 0; i < 64; ++i) s += expf(l[i * 65 + t] - m);
        lse = m + logf(s);
        for (int i = 0; i < 64; ++i) l[i * 65 + t] -= lse;
        __syncthreads();
    }
    for (int j = 0; j < 64; ++j) P[t * 64 + j] = expf(l[t * 65 + j]);
}

// ---------------- F (generalized gcd) -----------------------------------------
__device__ __forceinline__ float btln_clip01(float v) {
    return fminf(fmaxf(v, EPSF), 1.0f - EPSF);
}

// x,y must already be clipped; p = sqrt(3/max(2-a,eps)) - 1 precomputed for this a
__device__ __forceinline__ float btln_F_base(float x, float y, float a,
                                             float w0, float w1, float p) {
    float lin = w0 * x + w1 * y;
    // (x^(2w0) * y^(2w1))^p  ==  exp2(p * (2w0*log2 x + 2w1*log2 y))
    float g = exp2f(p * (2.0f * w0 * log2f(x) + 2.0f * w1 * log2f(y)));
    float rule0 = ((fabsf(x - 1.0f) < EPSF) && (fabsf(y - 1.0f) < EPSF)) ? 1.0f : 0.0f;
    float mid = (3.0f - 4.0f * a) * lin + (4.0f * a - 2.0f) * g;
    float r = (a >= 0.75f) ? g : ((a > 0.5f) ? mid : lin);
    return (fabsf(a - 2.0f) < EPSF) ? rule0 : r;
}

__device__ __forceinline__ float btln_F(float x, float y, float a, float w0, float w1,
                                        float p_hi, float p_lo) {
    x = btln_clip01(x);
    y = btln_clip01(y);
    float lin = w0 * x + w1 * y;
    float hi = btln_F_base(x, y, a, w0, w1, p_hi);
    float aref = fminf(fmaxf(1.0f - a, -1.0f + EPSF), 2.0f - EPSF);
    float lo = 1.0f - btln_F_base(btln_clip01(1.0f - x), btln_clip01(1.0f - y),
                                  aref, w0, w1, p_lo);
    float r = (fabsf(a - 0.5f) < EPSF) ? lin : ((a < 0.5f) ? lo : hi);
    return (r != r) ? a : r;  // NaN fallback to bias
}

// ---------------- Kernel 2: fused (x @ P) via f32 WMMA + sequential tree scan --
__global__ __launch_bounds__(256) void btln_fused_kernel(const float* __restrict__ x,
                                                         const float* __restrict__ P,
                                                         const float* __restrict__ weights,
                                                         const float* __restrict__ biases,
                                                         float* __restrict__ out) {
    __shared__ float PT[64 * 66];                  // P transposed: PT[n][k], pad 66 (8B aligned pairs)
    __shared__ float leaves[ROWS_PER_BLOCK * 66];  // leaves[local_row][leaf], pad 66
    __shared__ float pw0[NLAYERS], pw1[NLAYERS], pb[NLAYERS];
    __shared__ float pph[NLAYERS], ppl[NLAYERS];

    const int tid = threadIdx.x;

    // Stage P transposed into LDS (keeps B fragments as contiguous float2 in k)
    for (int e = tid; e < 64 * 64; e += 256) {
        int k = e >> 6, n = e & 63;
        PT[n * 66 + k] = P[e];
    }
    // Precompute per-layer scan constants
    if (tid < NLAYERS) {
        float w = weights[tid];
        float b = biases[tid];
        pw0[tid] = w;
        pw1[tid] = 1.0f - w;
        pb[tid]  = b;
        pph[tid] = sqrtf(3.0f / fmaxf(2.0f - b, EPSF)) - 1.0f;
        float aref = fminf(fmaxf(1.0f - b, -1.0f + EPSF), 2.0f - EPSF);
        ppl[tid] = sqrtf(3.0f / fmaxf(2.0f - aref, EPSF)) - 1.0f;
    }
    __syncthreads();

    const int wave = tid >> 5;
    const int lane = tid & 31;
    const int nn   = lane & 15;          // M for A-frag, N for B-frag (both use lane%16)
    const int kb   = (lane >> 4) * 2;    // half-wave selects K pair {0,1} vs {2,3}
    const long block_row0 = (long)blockIdx.x * ROWS_PER_BLOCK;
    const int lrow0 = wave * 16;         // this wave's 16-row m-tile

    // A fragments: lane reads its own x row (coalesced b64 per k-step)
    const float* arow = x + (block_row0 + lrow0 + nn) * 64 + kb;

    v8f acc0 = {}, acc1 = {}, acc2 = {}, acc3 = {};
#pragma unroll
    for (int s = 0; s < 16; ++s) {       // K = 64 in 16 steps of 4
        v2f a  = *(const v2f*)(arow + 4 * s);
        v2f b0 = *(const v2f*)&PT[( 0 + nn) * 66 + 4 * s + kb];
        v2f b1 = *(const v2f*)&PT[(16 + nn) * 66 + 4 * s + kb];
        v2f b2 = *(const v2f*)&PT[(32 + nn) * 66 + 4 * s + kb];
        v2f b3 = *(const v2f*)&PT[(48 + nn) * 66 + 4 * s + kb];
        acc0 = __builtin_amdgcn_wmma_f32_16x16x4_f32(false, a, false, b0, (short)0, acc0, false, false);
        acc1 = __builtin_amdgcn_wmma_f32_16x16x4_f32(false, a, false, b1, (short)0, acc1, false, false);
        acc2 = __builtin_amdgcn_wmma_f32_16x16x4_f32(false, a, false, b2, (short)0, acc2, false, false);
        acc3 = __builtin_amdgcn_wmma_f32_16x16x4_f32(false, a, false, b3, (short)0, acc3, false, false);
    }

    // Scatter D tiles to LDS: VGPR r -> M=r (lanes 0-15) / M=r+8 (lanes 16-31)
    const int mrow = lrow0 + ((lane >> 4) << 3);
#pragma unroll
    for (int r = 0; r < 8; ++r) {
        leaves[(mrow + r) * 66 +  0 + nn] = acc0[r];
        leaves[(mrow + r) * 66 + 16 + nn] = acc1[r];
        leaves[(mrow + r) * 66 + 32 + nn] = acc2[r];
        leaves[(mrow + r) * 66 + 48 + nn] = acc3[r];
    }
    __syncthreads();

    // Sequential left-leaning tree scan: one thread per batch row
    if (tid < ROWS_PER_BLOCK) {
        float st = leaves[tid * 66 + 0];
        for (int i = 0; i < NLAYERS; ++i) {
            float y = leaves[tid * 66 + i + 1];
            st = btln_F(st, y, pb[i], pw0[i], pw1[i], pph[i], ppl[i]);
        }
        out[block_row0 + tid] = st;
    }
}

extern "C" void kernel_launch(void* const* d_in, const int* in_sizes, int n_in,
                              void* d_out, int out_size, void* d_ws, size_t ws_size,
                              hipStream_t stream) {
    const float* x       = (const float*)d_in[0];
    const float* logits  = (const float*)d_in[1];
    const float* weights = (const float*)d_in[2];
    const float* biases  = (const float*)d_in[3];
    float* out = (float*)d_out;
    float* P   = (float*)d_ws;   // 64*64 floats of scratch

    btln_sinkhorn_kernel<<<1, 64, 0, stream>>>(logits, P);
    btln_fused_kernel<<<BATCH_TOTAL / ROWS_PER_BLOCK, 256, 0, stream>>>(x, P, weights, biases, out);
}